// ImagePaste_27650999451649
// MI455X (gfx1250) — compile-verified
//
#include <hip/hip_runtime.h>

// ImagePaste: B canvases [128,128,3] initialized to t_val; 3 sequential RGBA
// pastes (72x72) with first-write-wins gated on (alpha>0) && (canvas RGB sum
// == 765.0f). Memory-bound streaming kernel: 1 thread = 1 output pixel.
//
// Layouts (flat, row-major, matching the JAX reference):
//   images:    [B, 3, 72, 72, 4]  float32
//   positions: [B, 6] int32; paste p: dim0-offset = pos[2p+1], dim1-offset = pos[2p]
//   t_val:     [1] int32
//   out:       [B, 128, 128, 3] float32

#define IMG_SIZE   72
#define CANVAS     128
#define NUM_PASTES 3
#define WHITE_SUM  765.0f
#define PIX_PER_IMG (CANVAS * CANVAS)      // 16384
#define BLOCKS_PER_IMG (PIX_PER_IMG / 256) // 64 blocks of 256 threads (8 wave32)

typedef float v4f __attribute__((ext_vector_type(4)));  // native clang vector: NT-load legal

__global__ __launch_bounds__(256)
void ImagePaste_27650999451649_kernel(const float* __restrict__ images,
                                      const int*   __restrict__ positions,
                                      const int*   __restrict__ tval_p,
                                      float*       __restrict__ out) {
  // Each block covers 2 full canvas rows of exactly one batch image, so the
  // batch index and the 6 position ints are block-uniform -> scalarized s_loads.
  const int b   = (int)blockIdx.x >> 6;                           // blockIdx / BLOCKS_PER_IMG
  const int pix = (((int)blockIdx.x & (BLOCKS_PER_IMG - 1)) << 8) // 0..16383 within image
                  + (int)threadIdx.x;
  const int i = pix >> 7;     // canvas dim0
  const int j = pix & 127;    // canvas dim1

  const float tv = (float)tval_p[0];
  float r = tv, g = tv, bch = tv;
  float cansum = (tv + tv) + tv;   // matches jnp.sum reduction order over 3 elems

  const int pbase = b * 6;
  #pragma unroll
  for (int p = 0; p < NUM_PASTES; ++p) {
    const int off1 = positions[pbase + 2 * p];       // canvas dim1 offset (y0 in ref)
    const int off0 = positions[pbase + 2 * p + 1];   // canvas dim0 offset (x0 in ref)
    const int ri = i - off0;
    const int ci = j - off1;
    // Load only for lanes inside the tile that are still "white":
    // EXEC-masked lanes skip the 16B fetch entirely.
    if ((unsigned)ri < (unsigned)IMG_SIZE &&
        (unsigned)ci < (unsigned)IMG_SIZE &&
        cansum == WHITE_SUM) {
      const int eidx = ((b * NUM_PASTES + p) * IMG_SIZE + ri) * IMG_SIZE + ci;
      // 16B-aligned vec4 RGBA load, non-temporal (touch-once stream, bigger than L2).
      const v4f px = __builtin_nontemporal_load(
          reinterpret_cast<const v4f*>(images) + eidx);
      if (px.w > 0.0f) {
        r = px.x; g = px.y; bch = px.z;
        cansum = (r + g) + bch;   // later pastes gated on this sum == 765.0f
      }
    }
  }

  // [B,128,128,3] store: 3 adjacent dwords per lane, contiguous across the wave.
  float* o = out + (size_t)(b * PIX_PER_IMG + pix) * 3;
  __builtin_nontemporal_store(r,   o + 0);
  __builtin_nontemporal_store(g,   o + 1);
  __builtin_nontemporal_store(bch, o + 2);
}

extern "C" void kernel_launch(void* const* d_in, const int* in_sizes, int n_in,
                              void* d_out, int out_size, void* d_ws, size_t ws_size,
                              hipStream_t stream) {
  const float* images    = (const float*)d_in[0];
  const int*   positions = (const int*)d_in[1];
  const int*   tval      = (const int*)d_in[2];
  float*       out       = (float*)d_out;

  const int B = in_sizes[1] / 6;  // positions is [B,6]
  dim3 grid((unsigned)(B * BLOCKS_PER_IMG));
  dim3 block(256);
  ImagePaste_27650999451649_kernel<<<grid, block, 0, stream>>>(images, positions, tval, out);
}